// GNNEncoder_24867860643896
// MI455X (gfx1250) — compile-verified
//
#include <hip/hip_runtime.h>

#define BATCH   8
#define NNODES  10000
#define FEAT    128                    // F_IN == H == 128
#define NROWS   (BATCH * NNODES)       // 80000, divisible by 64
#define EPB     8                      // edges per wave in scatter kernel

typedef __attribute__((ext_vector_type(2))) float v2f;
typedef __attribute__((ext_vector_type(8))) float v8f;

// ---------------- degree / normalization ----------------

__global__ void k_deg_init(float* __restrict__ deg) {
    int i = blockIdx.x * blockDim.x + threadIdx.x;
    if (i < NNODES) deg[i] = 1.0f;               // self-loop contribution
}

__global__ void k_deg_count(const int* __restrict__ dst, float* __restrict__ deg, int nE) {
    int e = blockIdx.x * blockDim.x + threadIdx.x;
    if (e < nE) atomicAdd(&deg[dst[e]], 1.0f);
}

__global__ void k_dinv(const float* __restrict__ deg, float* __restrict__ dinv) {
    int i = blockIdx.x * blockDim.x + threadIdx.x;
    if (i < NNODES) dinv[i] = rsqrtf(deg[i]);    // deg >= 1 always
}

// ---------------- fused GEMM (fp32 WMMA) ----------------
// Per block: 4 waves, each wave owns one 16-row panel of A (all 128 K values
// held in registers, optional bias+ReLU applied at load) and produces the full
// 128-wide output row panel by looping over 8 column tiles.
// Epilogue writes BOTH:  outRaw = A@W   and   outScaled = (A@W) * dinv[n]^2
// (the self-loop term of the GCN aggregation), eliminating a separate pass.
// Because each row panel is read/written by exactly one wave, outScaled may
// alias the A input (used for layer 2 to keep workspace at 2 buffers).

template <bool RELU>
__global__ __launch_bounds__(128) void k_gemm_fused(const float* __restrict__ A,
                                                    const float* __restrict__ W,
                                                    const float* __restrict__ abias,
                                                    const float* __restrict__ dinv,
                                                    float* __restrict__ outRaw,
                                                    float* __restrict__ outScaled) {
    __shared__ float wsh[FEAT * FEAT];           // 64 KB: whole weight matrix
    {   // cooperative, coalesced stage of W into LDS
        const float4* wsrc = (const float4*)W;
        float4* wdst = (float4*)wsh;
        for (int i = threadIdx.x; i < (FEAT * FEAT) / 4; i += 128) wdst[i] = wsrc[i];
    }
    __syncthreads();

    const int lane = threadIdx.x & 31;
    const int wave = threadIdx.x >> 5;
    const int m0   = (blockIdx.x * 4 + wave) * 16;   // this wave's row-tile base
    const int mrow = m0 + (lane & 15);               // A: lane -> matrix row M
    const int koff = (lane >> 4) * 2;                // lanes 16-31 hold K+2,K+3
    const int nlo  = lane & 15;                      // B/C/D: lane -> col within tile

    // ---- load (and transform) this wave's full A panel into registers ----
    v2f afrag[FEAT / 4];
    const float* __restrict__ arow = A + (size_t)mrow * FEAT;
#pragma unroll
    for (int kk = 0; kk < FEAT / 4; ++kk) {
        v2f a = *(const v2f*)(arow + kk * 4 + koff);
        if (RELU) {
            a.x = fmaxf(a.x + abias[kk * 4 + koff + 0], 0.0f);
            a.y = fmaxf(a.y + abias[kk * 4 + koff + 1], 0.0f);
        }
        afrag[kk] = a;
    }

    // per-output-row self-loop scale dinv[n]^2 (row set is t-invariant)
    const int rbase = m0 + ((lane >> 4) << 3);       // C/D: M = v + 8*(lane>=16)
    float dsq[8];
#pragma unroll
    for (int v = 0; v < 8; ++v) {
        const float dd = dinv[(rbase + v) % NNODES];
        dsq[v] = dd * dd;
    }

    // ---- 8 column tiles, K-loop of fp32 WMMAs, B fragments from LDS ----
#pragma unroll 1
    for (int t = 0; t < 8; ++t) {
        const int n0 = t * 16;
        v8f c = {};
#pragma unroll
        for (int kk = 0; kk < FEAT / 4; ++kk) {
            const int kb = kk * 4 + koff;
            v2f b;
            b.x = wsh[(kb + 0) * FEAT + n0 + nlo];
            b.y = wsh[(kb + 1) * FEAT + n0 + nlo];
            c = __builtin_amdgcn_wmma_f32_16x16x4_f32(false, afrag[kk], false, b,
                                                      (short)0, c, false, false);
        }
#pragma unroll
        for (int v = 0; v < 8; ++v) {
            const size_t idx = (size_t)(rbase + v) * FEAT + n0 + nlo;
            outRaw[idx]    = c[v];
            outScaled[idx] = c[v] * dsq[v];
        }
    }
}

// ---------------- edge scatter:  out[b,dst,:] += h[b,src,:] * dinv[src]*dinv[dst] ----
// one wave per edge: lane gathers float4 (coalesced 512B/wave), 4 dword atomics.
// src/dst/dinv are wave-uniform -> scalar loads.

__global__ __launch_bounds__(128) void k_agg_edges(const float* __restrict__ h,
                                                   const int* __restrict__ src,
                                                   const int* __restrict__ dst,
                                                   const float* __restrict__ dinv,
                                                   float* __restrict__ out, int nE) {
    const int lane = threadIdx.x & 31;
    const int wv   = threadIdx.x >> 5;
    const int c4   = lane * 4;
    const size_t bbase = (size_t)blockIdx.y * NNODES * FEAT;
    const int e0 = (blockIdx.x * 4 + wv) * EPB;
#pragma unroll
    for (int i = 0; i < EPB; ++i) {
        const int e = e0 + i;
        if (e >= nE) break;
        const int s = src[e];
        const int d = dst[e];
        const float w = dinv[s] * dinv[d];
        const float4 hv = *(const float4*)(h + bbase + (size_t)s * FEAT + c4);
        float* __restrict__ op = out + bbase + (size_t)d * FEAT + c4;
        atomicAdd(op + 0, hv.x * w);
        atomicAdd(op + 1, hv.y * w);
        atomicAdd(op + 2, hv.z * w);
        atomicAdd(op + 3, hv.w * w);
    }
}

// ---------------- LayerNorm(h + b2) * ln_w + ln_b ----------------
// one wave32 per row; each lane owns 4 contiguous features (float4)

__global__ __launch_bounds__(256) void k_layernorm(const float* __restrict__ h,
                                                   const float* __restrict__ bias,
                                                   const float* __restrict__ lnw,
                                                   const float* __restrict__ lnb,
                                                   float* __restrict__ out) {
    const int lane = threadIdx.x & 31;
    const int wv = threadIdx.x >> 5;
    const size_t row = (size_t)blockIdx.x * 8 + wv;   // 10000 blocks * 8 waves

    const float4 hv = ((const float4*)(h + row * FEAT))[lane];
    const float4 bv = ((const float4*)bias)[lane];
    float4 v = make_float4(hv.x + bv.x, hv.y + bv.y, hv.z + bv.z, hv.w + bv.w);

    float s = v.x + v.y + v.z + v.w;
#pragma unroll
    for (int m = 16; m >= 1; m >>= 1) s += __shfl_xor(s, m, 32);
    const float mu = s * (1.0f / FEAT);

    const float dx = v.x - mu, dy = v.y - mu, dz = v.z - mu, dw = v.w - mu;
    float q = dx * dx + dy * dy + dz * dz + dw * dw;
#pragma unroll
    for (int m = 16; m >= 1; m >>= 1) q += __shfl_xor(q, m, 32);
    const float r = rsqrtf(q * (1.0f / FEAT) + 1e-5f);

    const float4 wv4 = ((const float4*)lnw)[lane];
    const float4 bv4 = ((const float4*)lnb)[lane];
    float4 o;
    o.x = dx * r * wv4.x + bv4.x;
    o.y = dy * r * wv4.y + bv4.y;
    o.z = dz * r * wv4.z + bv4.z;
    o.w = dw * r * wv4.w + bv4.w;
    ((float4*)(out + row * FEAT))[lane] = o;
}

// ---------------- launch ----------------

extern "C" void kernel_launch(void* const* d_in, const int* in_sizes, int n_in,
                              void* d_out, int out_size, void* d_ws, size_t ws_size,
                              hipStream_t stream) {
    const float* x    = (const float*)d_in[0];
    const int*   ei   = (const int*)d_in[1];      // [2, E]
    const float* W1   = (const float*)d_in[2];
    const float* b1   = (const float*)d_in[3];
    const float* W2   = (const float*)d_in[4];
    const float* b2   = (const float*)d_in[5];
    const float* ln_w = (const float*)d_in[6];
    const float* ln_b = (const float*)d_in[7];

    const int E = in_sizes[1] / 2;
    const int* src = ei;
    const int* dst = ei + E;

    // workspace layout: deg | dinv | buf0 | buf1
    float* deg  = (float*)d_ws;                  // NNODES
    float* dinv = deg + NNODES;                  // NNODES
    size_t off = ((size_t)2 * NNODES * sizeof(float) + 255) & ~(size_t)255;
    float* buf0 = (float*)((char*)d_ws + off);              // NROWS*FEAT
    float* buf1 = buf0 + (size_t)NROWS * FEAT;              // NROWS*FEAT

    const int gemmBlocks = NROWS / 64;           // 1250 blocks x 4 waves x 16 rows
    const dim3 aggGrid((E + 4 * EPB - 1) / (4 * EPB), BATCH);

    // normalization coefficients (graph shared across batch)
    k_deg_init<<<(NNODES + 255) / 256, 256, 0, stream>>>(deg);
    k_deg_count<<<(E + 255) / 256, 256, 0, stream>>>(dst, deg, E);
    k_dinv<<<(NNODES + 255) / 256, 256, 0, stream>>>(deg, dinv);

    // layer 1: T1 = X@W1 -> buf0 (raw), buf1 = T1*dinv^2 (self-loop init)
    k_gemm_fused<false><<<gemmBlocks, 128, 0, stream>>>(x, W1, nullptr, dinv, buf0, buf1);
    // buf1 += scatter of buf0 over edges  => A1 (pre-bias, pre-ReLU)
    k_agg_edges<<<aggGrid, 128, 0, stream>>>(buf0, src, dst, dinv, buf1, E);

    // layer 2: A-load applies relu(A1 + b1); outScaled aliases A in-place (safe:
    // each wave registers its whole row panel before its epilogue stores)
    k_gemm_fused<true><<<gemmBlocks, 128, 0, stream>>>(buf1, W2, b1, dinv, buf0, buf1);
    // buf1 += scatter of buf0 over edges  => A2
    k_agg_edges<<<aggGrid, 128, 0, stream>>>(buf0, src, dst, dinv, buf1, E);

    // out = LayerNorm(A2 + b2)
    k_layernorm<<<NROWS / 8, 256, 0, stream>>>(buf1, b2, ln_w, ln_b, (float*)d_out);
}